// QConv2d_36086315221323
// MI455X (gfx1250) — compile-verified
//
#include <hip/hip_runtime.h>

typedef __attribute__((ext_vector_type(2))) float v2f;
typedef __attribute__((ext_vector_type(8))) float v8f;

#define CIN   32
#define COUT  32
#define HH    224
#define WW    224
#define KPAIRS 144       // (9*32)/2
#define TILE_W 32        // output pixels per block (ow)
#define XT_W   34        // TILE_W + 2 halo

__global__ __launch_bounds__(128) void QConv2d_36086315221323_kernel(
    const float* __restrict__ x,      // [16,32,224,224]
    const float* __restrict__ wgt,    // [32,32,3,3]
    const float* __restrict__ bias,   // [32]
    float* __restrict__ out)          // [16,32,224,224]
{
    // LDS: weights as channel-pair interleaved GEMM-A, input tile as GEMM-B source
    __shared__ float Wl[KPAIRS][COUT][2];     // [kpair][cout][k&1]  36 KB
    __shared__ float Xl[CIN/2][3][XT_W][2];   // [cpair][row][col][c&1] ~12.75 KB
    __shared__ float biasl[COUT];

    const int tid   = threadIdx.x;
    const int lane  = tid & 31;
    const int wave  = tid >> 5;          // 0..3
    const int chunk = blockIdx.x;        // 0..6
    const int oh    = blockIdx.y;        // 0..223
    const int b     = blockIdx.z;        // 0..15
    const int ow0   = chunk * TILE_W;

    // ---- stage weights: k = (r*3+s)*32 + cin, pair-interleaved for b64 loads ----
    for (int idx = tid; idx < COUT * CIN * 9; idx += 128) {
        int cout = idx / (CIN * 9);
        int rem  = idx - cout * (CIN * 9);
        int cin  = rem / 9;
        int rs   = rem - cin * 9;
        int k    = rs * 32 + cin;
        Wl[k >> 1][cout][k & 1] = wgt[idx];
    }
    if (tid < COUT) biasl[tid] = bias[tid];

    // ---- stage input tile with zero-padded halo ----
    for (int idx = tid; idx < CIN * 3 * XT_W; idx += 128) {
        int c   = idx / (3 * XT_W);
        int rem = idx - c * (3 * XT_W);
        int r   = rem / XT_W;
        int col = rem - r * XT_W;
        int ih = oh + r - 1;
        int iw = ow0 + col - 1;
        float v = 0.0f;
        if (ih >= 0 && ih < HH && iw >= 0 && iw < WW)
            v = x[(((long)b * CIN + c) * HH + ih) * WW + iw];
        Xl[c >> 1][r][col][c & 1] = v;
    }
    __syncthreads();

    // wave tile: 16 couts x 16 pixels
    const int cout_base = (wave & 1) * 16;
    const int noff      = (wave >> 1) * 16;
    const int m         = cout_base + (lane & 15);  // A-matrix row (M)
    const int col       = noff + (lane & 15);       // B-matrix col (N) in tile
    const int lhalf     = lane >> 4;                // 0: K=k0,k0+1  1: K=k0+2,k0+3

    v8f acc0 = {}; v8f acc1 = {};

#pragma unroll
    for (int rs = 0; rs < 9; ++rs) {
        const int r = rs / 3, s = rs % 3;
#pragma unroll
        for (int c0 = 0; c0 < CIN; c0 += 8) {
            {   // K-step: channels c0 .. c0+3
                int cp = (c0 >> 1) + lhalf;
                v2f a  = *(const v2f*)&Wl[rs * 16 + cp][m][0];
                v2f bb = *(const v2f*)&Xl[cp][r][col + s][0];
                acc0 = __builtin_amdgcn_wmma_f32_16x16x4_f32(
                    false, a, false, bb, (short)0, acc0, false, false);
            }
            {   // K-step: channels c0+4 .. c0+7
                int cp = ((c0 + 4) >> 1) + lhalf;
                v2f a  = *(const v2f*)&Wl[rs * 16 + cp][m][0];
                v2f bb = *(const v2f*)&Xl[cp][r][col + s][0];
                acc1 = __builtin_amdgcn_wmma_f32_16x16x4_f32(
                    false, a, false, bb, (short)0, acc1, false, false);
            }
        }
    }
    v8f acc = acc0 + acc1;

    // ---- epilogue: C/D layout -> NCHW store with bias ----
    const int ow = ow0 + col;
#pragma unroll
    for (int j = 0; j < 8; ++j) {
        int cout = cout_base + j + (lhalf << 3);
        out[(((long)b * COUT + cout) * HH + oh) * WW + ow] = acc[j] + biasl[cout];
    }
}

extern "C" void kernel_launch(void* const* d_in, const int* in_sizes, int n_in,
                              void* d_out, int out_size, void* d_ws, size_t ws_size,
                              hipStream_t stream) {
    (void)in_sizes; (void)n_in; (void)d_ws; (void)ws_size; (void)out_size;
    const float* x    = (const float*)d_in[0];
    const float* wgt  = (const float*)d_in[1];
    const float* bias = (const float*)d_in[2];
    float* out        = (float*)d_out;

    dim3 grid(WW / TILE_W, HH, 16);   // 7 x 224 x 16
    dim3 block(128);
    QConv2d_36086315221323_kernel<<<grid, block, 0, stream>>>(x, wgt, bias, out);
}